// LagAwarePeerBlock_80487687127554
// MI455X (gfx1250) — compile-verified
//
#include <hip/hip_runtime.h>
#include <hip/hip_bf16.h>
#include <math.h>

// ---------------------------------------------------------------------------
// Problem constants (B=8, N=64, T=256, H=128, L=5, TOP_K=3)
// ---------------------------------------------------------------------------
#define Bdim   8
#define Npeer  64
#define Tdim   256
#define Hdim   128
#define Llags  5
#define NL     (Npeer * Llags)   // 320
#define TOPK   3
#define LN_EPS 1e-5f

typedef float v2f __attribute__((ext_vector_type(2)));
typedef float v8f __attribute__((ext_vector_type(8)));

// ---------------------------------------------------------------------------
// WMMA f32 GEMM:  Y[M,128] = act( X[M,128] @ op(W) + bias ),  M = gridDim.x*128
//   transB=1: op(W)[k][n] = W[n][k]   (i.e. X @ W^T)
//   transB=0: op(W)[k][n] = W[k][n]   (i.e. X @ W)
//   act: 0 = none, 1 = ELU(alpha=1)
// Block: 256 threads = 8 waves. Wave w computes rows [w*16, w*16+16) x all 128
// cols as 8 accumulators of v8f, K staged through LDS in chunks of 32.
// ---------------------------------------------------------------------------
#define KCHUNK 32
#define LPAD   36   // KCHUNK + 4 -> conflict-free f32x2 lane reads

__global__ __launch_bounds__(256) void gemm128_kernel(
    const float* __restrict__ X, const float* __restrict__ W,
    const float* __restrict__ bias, float* __restrict__ Y,
    int transB, int act)
{
    __shared__ float Xs[128 * LPAD];
    __shared__ float Bs[128 * LPAD];

    const int tid   = threadIdx.x;
    const int wave  = tid >> 5;
    const int lane  = tid & 31;
    const int lrow  = lane & 15;      // row/col-in-tile index
    const int khalf = lane >> 4;      // 0: K={0,1}, 1: K={2,3}
    const long mBase = (long)blockIdx.x * 128;

    v8f acc[8];
    #pragma unroll
    for (int i = 0; i < 8; ++i) {
        #pragma unroll
        for (int j = 0; j < 8; ++j) acc[i][j] = 0.0f;
    }

    for (int kc = 0; kc < Hdim / KCHUNK; ++kc) {
        const int k0 = kc * KCHUNK;

        // Stage X rows [mBase..+128), cols [k0..k0+32)  (coalesced float4)
        #pragma unroll
        for (int it = 0; it < 4; ++it) {
            int li  = it * 256 + tid;     // 0..1023
            int row = li >> 3;            // 0..127
            int c4  = li & 7;             // 0..7
            float4 v = *(const float4*)(X + (mBase + row) * Hdim + k0 + c4 * 4);
            *(float4*)(&Xs[row * LPAD + c4 * 4]) = v;
        }
        // Stage op(W): Bs[n][c] = op(W)[k0+c][n]
        if (transB) {
            #pragma unroll
            for (int it = 0; it < 4; ++it) {
                int li = it * 256 + tid;
                int n  = li >> 3;
                int c4 = li & 7;
                float4 v = *(const float4*)(W + (long)n * Hdim + k0 + c4 * 4);
                *(float4*)(&Bs[n * LPAD + c4 * 4]) = v;
            }
        } else {
            #pragma unroll
            for (int it = 0; it < 4; ++it) {
                int li = it * 256 + tid;
                int kr = li >> 5;             // 0..31 (k within chunk)
                int n4 = (li & 31) * 4;       // col group
                float4 v = *(const float4*)(W + (long)(k0 + kr) * Hdim + n4);
                Bs[(n4 + 0) * LPAD + kr] = v.x;
                Bs[(n4 + 1) * LPAD + kr] = v.y;
                Bs[(n4 + 2) * LPAD + kr] = v.z;
                Bs[(n4 + 3) * LPAD + kr] = v.w;
            }
        }
        __syncthreads();

        // Compute: 8 k-steps of 4 x 8 n-tiles
        const float* xrow = &Xs[(wave * 16 + lrow) * LPAD + khalf * 2];
        #pragma unroll
        for (int kk = 0; kk < KCHUNK / 4; ++kk) {
            v2f a;
            a[0] = xrow[kk * 4 + 0];
            a[1] = xrow[kk * 4 + 1];
            #pragma unroll
            for (int nt = 0; nt < 8; ++nt) {
                const float* brow = &Bs[(nt * 16 + lrow) * LPAD + kk * 4 + khalf * 2];
                v2f bf;
                bf[0] = brow[0];
                bf[1] = brow[1];
                acc[nt] = __builtin_amdgcn_wmma_f32_16x16x4_f32(
                    false, a, false, bf, (short)0, acc[nt], false, false);
            }
        }
        __syncthreads();
    }

    // Epilogue: bias + activation + store.
    // C layout: VGPR j -> row = j + khalf*8, col = lrow (within tile)
    #pragma unroll
    for (int nt = 0; nt < 8; ++nt) {
        int col = nt * 16 + lrow;
        float bb = bias ? bias[col] : 0.0f;
        #pragma unroll
        for (int j = 0; j < 8; ++j) {
            long row = mBase + wave * 16 + j + khalf * 8;
            float v = acc[nt][j] + bb;
            if (act == 1) v = (v > 0.0f) ? v : expm1f(v);
            Y[row * Hdim + col] = v;
        }
    }
}

// ---------------------------------------------------------------------------
// Wave (32-lane) reduction
// ---------------------------------------------------------------------------
__device__ __forceinline__ float waveReduceSum(float v) {
    #pragma unroll
    for (int o = 16; o > 0; o >>= 1) v += __shfl_xor(v, o, 32);
    return v;
}

// ---------------------------------------------------------------------------
// Attention: per (b,t) block.
//   logits[n,l] = (Qk[b,t] . peer_h[b,n,t-lag_l] + Q.bk) / sqrt(H)
//   top-3 -> softmax -> blend raw rows -> cs_y = Wv . z + (sum w) * bv
// ---------------------------------------------------------------------------
__global__ __launch_bounds__(256) void attn_kernel(
    const float* __restrict__ Q,          // [B*T,128]
    const float* __restrict__ Qk,         // [B*T,128]
    const float* __restrict__ peer_h,     // [B,N,T,H]
    const unsigned char* __restrict__ peer_mask, // [B,N] (bool)
    const float* __restrict__ bk,         // [128]
    const float* __restrict__ Wv,         // [128,128]
    const float* __restrict__ bvv,        // [128]
    float* __restrict__ csy)              // [B*T,128]
{
    __shared__ float qk_s[Hdim];
    __shared__ float logit_s[NL];
    __shared__ int   sel_n[TOPK];
    __shared__ int   sel_tl[TOPK];
    __shared__ float sel_w[TOPK];
    __shared__ float wsum_s;
    __shared__ float qb_s;
    __shared__ float z_s[Hdim];

    const int bt   = blockIdx.x;
    const int b    = bt >> 8;      // T = 256
    const int t    = bt & 255;
    const int tid  = threadIdx.x;
    const int wave = tid >> 5;
    const int lane = tid & 31;
    const int lags[Llags] = {1, 5, 10, 21, 30};

    if (tid < Hdim) qk_s[tid] = Qk[(long)bt * Hdim + tid];

    if (wave == 0) {  // qb = Q[b,t] . bk
        float4 q4 = *(const float4*)(Q + (long)bt * Hdim + lane * 4);
        float4 k4 = *(const float4*)(bk + lane * 4);
        float s = q4.x * k4.x + q4.y * k4.y + q4.z * k4.z + q4.w * k4.w;
        s = waveReduceSum(s);
        if (lane == 0) qb_s = s;
    }
    __syncthreads();

    const float scale = 0.08838834764831845f;  // 1/sqrt(128)
    float4 qk4 = *(const float4*)(&qk_s[lane * 4]);

    // 320 logits, 8 waves, one wave-cooperative dot per (n,l)
    for (int p = wave; p < NL; p += 8) {
        int n  = p / Llags;
        int l  = p - n * Llags;            // flat index = n*L + l (matches reshape)
        int tl = t - lags[l];
        bool valid = (tl >= 0) && (peer_mask[b * Npeer + n] != 0);
        int tlc = tl < 0 ? 0 : tl;
        const float* pr = peer_h + (((long)(b * Npeer + n)) * Tdim + tlc) * Hdim;
        float4 x4 = *(const float4*)(pr + lane * 4);
        float s = x4.x * qk4.x + x4.y * qk4.y + x4.z * qk4.z + x4.w * qk4.w;
        s = waveReduceSum(s);
        if (lane == 0) logit_s[p] = valid ? (s + qb_s) * scale : -INFINITY;
    }
    __syncthreads();

    // top-3 (3 passes of wave-parallel argmax; tie -> lowest index, like top_k)
    if (wave == 0) {
        float vals[TOPK];
        int   idxs[TOPK];
        for (int j = 0; j < TOPK; ++j) {
            float bvl = -INFINITY;
            int   bil = NL;
            for (int p = lane; p < NL; p += 32) {
                float v = logit_s[p];
                if (v > bvl || (v == bvl && p < bil)) { bvl = v; bil = p; }
            }
            #pragma unroll
            for (int o = 16; o > 0; o >>= 1) {
                float ov = __shfl_xor(bvl, o, 32);
                int   oi = __shfl_xor(bil, o, 32);
                if (ov > bvl || (ov == bvl && oi < bil)) { bvl = ov; bil = oi; }
            }
            vals[j] = bvl;
            idxs[j] = bil;
            if (lane == 0 && bil < NL) logit_s[bil] = -INFINITY;  // exclude; in-order LDS
        }
        if (lane == 0) {
            bool all_inf = (vals[0] == -INFINITY);
            float sv[TOPK], m = -3.0e38f;
            for (int j = 0; j < TOPK; ++j) {
                sv[j] = (vals[j] == -INFINITY) ? -1e9f : vals[j];
                m = fmaxf(m, sv[j]);
            }
            float e[TOPK], se = 0.0f;
            for (int j = 0; j < TOPK; ++j) { e[j] = expf(sv[j] - m); se += e[j]; }
            for (int j = 0; j < TOPK; ++j) {
                sel_w[j] = all_inf ? 0.0f : (e[j] / se);
                int p = (idxs[j] < NL) ? idxs[j] : 0;
                int n = p / Llags;
                int l = p - n * Llags;
                sel_n[j] = n;
                int tl = t - lags[l];
                sel_tl[j] = tl < 0 ? 0 : tl;
            }
            wsum_s = all_inf ? 0.0f : 1.0f;
        }
    }
    __syncthreads();

    // z = sum_j w_j * peer_h[b, n_j, tl_j, :]
    if (tid < Hdim) {
        float z = 0.0f;
        #pragma unroll
        for (int j = 0; j < TOPK; ++j) {
            const float* pr = peer_h + (((long)(b * Npeer + sel_n[j])) * Tdim + sel_tl[j]) * Hdim;
            z += sel_w[j] * pr[tid];
        }
        z_s[tid] = z;
    }
    __syncthreads();

    // cs_y = Wv . z + wsum * bv   (wave w handles h = w*16 .. w*16+15)
    float4 zz = *(const float4*)(&z_s[lane * 4]);
    for (int hh = 0; hh < 16; ++hh) {
        int h = wave * 16 + hh;
        float4 w4 = *(const float4*)(Wv + (long)h * Hdim + lane * 4);
        float s = w4.x * zz.x + w4.y * zz.y + w4.z * zz.z + w4.w * zz.w;
        s = waveReduceSum(s);
        if (lane == 0) csy[(long)bt * Hdim + h] = s + wsum_s * bvv[h];
    }
}

// ---------------------------------------------------------------------------
// Residual + LayerNorm: out = LN(csy + F) * gamma + beta, one row per block
// ---------------------------------------------------------------------------
__global__ __launch_bounds__(128) void ln_kernel(
    const float* __restrict__ csy, const float* __restrict__ F,
    const float* __restrict__ gamma, const float* __restrict__ beta,
    float* __restrict__ out)
{
    __shared__ float red[4];
    const int r = blockIdx.x, tid = threadIdx.x;
    const int wave = tid >> 5, lane = tid & 31;

    float x = csy[(long)r * Hdim + tid] + F[(long)r * Hdim + tid];

    float s = waveReduceSum(x);
    if (lane == 0) red[wave] = s;
    __syncthreads();
    float mu = (red[0] + red[1] + red[2] + red[3]) * (1.0f / Hdim);
    __syncthreads();

    float d = x - mu;
    float s2 = waveReduceSum(d * d);
    if (lane == 0) red[wave] = s2;
    __syncthreads();
    float var = (red[0] + red[1] + red[2] + red[3]) * (1.0f / Hdim);

    out[(long)r * Hdim + tid] = d * rsqrtf(var + LN_EPS) * gamma[tid] + beta[tid];
}

// ---------------------------------------------------------------------------
// Launch
// ---------------------------------------------------------------------------
extern "C" void kernel_launch(void* const* d_in, const int* in_sizes, int n_in,
                              void* d_out, int out_size, void* d_ws, size_t ws_size,
                              hipStream_t stream) {
    const float* target_h = (const float*)d_in[0];   // [B,T,H]
    const float* peer_h   = (const float*)d_in[1];   // [B,N,T,H]
    const unsigned char* peer_mask = (const unsigned char*)d_in[2]; // [B,N] bool
    const float* Wq = (const float*)d_in[3];
    const float* bq = (const float*)d_in[4];
    const float* Wk = (const float*)d_in[5];
    const float* bk = (const float*)d_in[6];
    const float* Wv = (const float*)d_in[7];
    const float* bv = (const float*)d_in[8];
    const float* W1 = (const float*)d_in[9];
    const float* b1 = (const float*)d_in[10];
    const float* W2 = (const float*)d_in[11];
    const float* b2 = (const float*)d_in[12];
    const float* gamma = (const float*)d_in[13];
    const float* beta  = (const float*)d_in[14];

    const int M = Bdim * Tdim;                 // 2048
    float* ws  = (float*)d_ws;
    float* Qb  = ws;                           // M*128
    float* Qkb = ws + 1 * (size_t)M * Hdim;    // M*128
    float* Cy  = ws + 2 * (size_t)M * Hdim;    // cs_y
    float* H1  = ws + 3 * (size_t)M * Hdim;    // ffn hidden
    float* Fo  = ws + 4 * (size_t)M * Hdim;    // ffn out
    float* out = (float*)d_out;

    dim3 blk(256);
    const int gb = M / 128;                    // 16 blocks per GEMM

    // Q = target_h @ Wq^T + bq
    gemm128_kernel<<<gb, blk, 0, stream>>>(target_h, Wq, bq, Qb, 1, 0);
    // Qk = Q @ Wk   (folds the K projection into the query side)
    gemm128_kernel<<<gb, blk, 0, stream>>>(Qb, Wk, nullptr, Qkb, 0, 0);
    // attention: logits vs raw peer rows, top-3, blend, Wv matvec
    attn_kernel<<<M, blk, 0, stream>>>(Qb, Qkb, peer_h, peer_mask, bk, Wv, bv, Cy);
    // FFN
    gemm128_kernel<<<gb, blk, 0, stream>>>(Cy, W1, b1, H1, 1, 1);   // ELU
    gemm128_kernel<<<gb, blk, 0, stream>>>(H1, W2, b2, Fo, 1, 0);
    // residual + LayerNorm
    ln_kernel<<<M, dim3(128), 0, stream>>>(Cy, Fo, gamma, beta, out);
}